// StructuredLinearBlocks_69380901700091
// MI455X (gfx1250) — compile-verified
//
#include <hip/hip_runtime.h>
#include <hip/hip_bf16.h>

typedef __attribute__((ext_vector_type(2))) float v2f;
typedef __attribute__((ext_vector_type(4))) float v4f;
typedef __attribute__((ext_vector_type(8))) float v8f;
typedef __attribute__((ext_vector_type(4))) int   v4i;

typedef __attribute__((address_space(1))) v4i* as1_v4i_ptr;   // global
typedef __attribute__((address_space(3))) v4i* as3_v4i_ptr;   // LDS

#define BATCH   8192
#define IN_F    4096
#define OUT_F   4096
#define NBLK    64
#define BLK     64      // src == dst == 64
#define M_TILE  128     // batch rows per workgroup
#define LSTRIDE 68      // padded row stride (dwords): 16B-aligned, bank-conflict free

#if __has_builtin(__builtin_amdgcn_global_load_async_to_lds_b128)
#define USE_ASYNC_LDS 1
#else
#define USE_ASYNC_LDS 0
#endif

// Copy 16 bytes global -> LDS. Async path uses GLOBAL_LOAD_ASYNC_TO_LDS_B128
// (ASYNCcnt, no VGPR staging); fallback stages through a v4f register.
__device__ __forceinline__ void g2l_b128(const float* __restrict__ gsrc,
                                         float* __restrict__ ldst) {
#if USE_ASYNC_LDS
    __builtin_amdgcn_global_load_async_to_lds_b128(
        (as1_v4i_ptr)gsrc, (as3_v4i_ptr)ldst, /*offset=*/0, /*cpol=*/0);
#else
    *(v4f*)ldst = __builtin_nontemporal_load((const v4f*)gsrc);
#endif
}

__device__ __forceinline__ void wait_g2l() {
#if USE_ASYNC_LDS
#if __has_builtin(__builtin_amdgcn_s_wait_asynccnt)
    __builtin_amdgcn_s_wait_asynccnt(0);
#else
    asm volatile("s_wait_asynccnt 0" ::: "memory");
#endif
#endif
}

__global__ __launch_bounds__(256)
void StructuredLinearBlocks_69380901700091_kernel(const float* __restrict__ x,
                                                  const float* __restrict__ blocks,
                                                  float* __restrict__ out)
{
    __shared__ float lds_x[M_TILE * LSTRIDE];  // 128 x 64 x-tile (padded)
    __shared__ float lds_w[BLK * LSTRIDE];     // 64 x 64 weight block (padded)

    const int n    = blockIdx.y;            // diagonal block index
    const int row0 = blockIdx.x * M_TILE;   // batch row base
    const int tid  = threadIdx.x;           // 0..255

    // ---- stage weight block n into LDS: 64x64 f32 = 1024 float4, 4 per thread ----
    {
        const float* wsrc = blocks + (size_t)n * BLK * BLK;
        #pragma unroll
        for (int i = 0; i < 4; ++i) {
            int idx = tid + 256 * i;        // float4 index 0..1023
            int r   = idx >> 4;             // row (dst) 0..63
            int c4  = idx & 15;             // float4 column
            g2l_b128(wsrc + r * BLK + c4 * 4, &lds_w[r * LSTRIDE + c4 * 4]);
        }
    }
    // ---- stage x tile into LDS: 128x64 f32 = 2048 float4, 8 per thread ----
    {
        const float* xsrc = x + (size_t)row0 * IN_F + n * BLK;
        #pragma unroll
        for (int i = 0; i < 8; ++i) {
            int idx = tid + 256 * i;        // 0..2047
            int r   = idx >> 4;             // tile row 0..127
            int c4  = idx & 15;
            g2l_b128(xsrc + (size_t)r * IN_F + c4 * 4, &lds_x[r * LSTRIDE + c4 * 4]);
        }
    }
    wait_g2l();          // this wave's async copies complete
    __syncthreads();     // all waves' LDS writes visible

    // wave32: 8 waves, each owns 16 batch rows x all 64 output cols
    const int wave  = tid >> 5;             // 0..7
    const int lane  = tid & 31;
    const int lo    = lane & 15;            // row-within-fragment / col
    const int hi    = lane >> 4;            // selects K pair (A/B) or +8 rows (C)
    const int mbase = wave * 16;

    v8f c0 = {}, c1 = {}, c2 = {}, c3 = {};

    // K = 64 -> 16 steps of 16x16x4 f32 WMMA, 4 n-tiles of 16 cols
    #pragma unroll
    for (int kk = 0; kk < 16; ++kk) {
        const int k0 = kk * 4 + 2 * hi;
        // A (16x4): lane lo = row, VGPR pair = K {k0, k0+1} (hi selects K+2)
        v2f a  = *(const v2f*)(&lds_x[(mbase + lo) * LSTRIDE + k0]);
        // B (4x16): B[k][col] = W[col][k]; lane lo = col within n-tile
        v2f b0 = *(const v2f*)(&lds_w[( 0 + lo) * LSTRIDE + k0]);
        v2f b1 = *(const v2f*)(&lds_w[(16 + lo) * LSTRIDE + k0]);
        v2f b2 = *(const v2f*)(&lds_w[(32 + lo) * LSTRIDE + k0]);
        v2f b3 = *(const v2f*)(&lds_w[(48 + lo) * LSTRIDE + k0]);
        // (neg_a, A, neg_b, B, c_mod, C, reuse_a, reuse_b)
        c0 = __builtin_amdgcn_wmma_f32_16x16x4_f32(false, a, false, b0, (short)0, c0, false, false);
        c1 = __builtin_amdgcn_wmma_f32_16x16x4_f32(false, a, false, b1, (short)0, c1, false, false);
        c2 = __builtin_amdgcn_wmma_f32_16x16x4_f32(false, a, false, b2, (short)0, c2, false, false);
        c3 = __builtin_amdgcn_wmma_f32_16x16x4_f32(false, a, false, b3, (short)0, c3, false, false);
    }

    // ---- store 16x64 stripe: C VGPR r -> row r (lanes 0-15) / row r+8 (lanes 16-31)
    // y is write-once: non-temporal stores keep the 134 MB write stream out of L2.
    float* obase = out + (size_t)(row0 + mbase) * OUT_F + n * BLK;
    #pragma unroll
    for (int r = 0; r < 8; ++r) {
        float* orow = obase + (size_t)(r + 8 * hi) * OUT_F;
        __builtin_nontemporal_store(c0[r], &orow[ 0 + lo]);
        __builtin_nontemporal_store(c1[r], &orow[16 + lo]);
        __builtin_nontemporal_store(c2[r], &orow[32 + lo]);
        __builtin_nontemporal_store(c3[r], &orow[48 + lo]);
    }
}

extern "C" void kernel_launch(void* const* d_in, const int* in_sizes, int n_in,
                              void* d_out, int out_size, void* d_ws, size_t ws_size,
                              hipStream_t stream) {
    const float* x      = (const float*)d_in[0];   // [8192, 4096] f32
    const float* blocks = (const float*)d_in[1];   // [64, 64, 64]  f32
    float* out          = (float*)d_out;           // [8192, 4096] f32

    dim3 grid(BATCH / M_TILE, NBLK);               // (64, 64)
    dim3 block(256);
    StructuredLinearBlocks_69380901700091_kernel<<<grid, block, 0, stream>>>(x, blocks, out);
}